// Linear_36163624632513
// MI455X (gfx1250) — compile-verified
//
#include <hip/hip_runtime.h>

typedef float v2f __attribute__((ext_vector_type(2)));
typedef float v8f __attribute__((ext_vector_type(8)));

#define M_TOTAL 262144
#define KDIM 128
#define NDIM 128

// Each block: 8 waves * 16 rows = 128 rows of x.
// W (128x128 f32, 64KB) lives in LDS, pre-swizzled into V_WMMA_F32_16X16X4_F32
// B-fragment layout so each fragment is one contiguous ds_load_b64 per lane.
__global__ __launch_bounds__(256) void linear_wmma_f32_kernel(
    const float* __restrict__ x,      // [M, 128]
    const float* __restrict__ w,      // [128, 128] (K-major rows: w[k*128 + n])
    const float* __restrict__ bias,   // [128]
    float* __restrict__ y)            // [M, 128]
{
    // Swizzled W: fragment f = kk*8 + nt holds the 4x16 B tile (K=4kk..4kk+3,
    // N=nt*16..nt*16+15) as 32 float2 entries indexed by lane (slot stride 64 float2).
    __shared__ float ldsW[KDIM * NDIM];   // 64 KB

    const int t = threadIdx.x;

    // ---- Phase 1: cooperative load + swizzle of W into LDS ----
    // B-fragment layout (ISA 7.12.2, 32-bit B 4x16):
    //   lane l (<16):  v0 = W[4kk+0][n0+l],   v1 = W[4kk+1][n0+l]
    //   lane l (>=16): v0 = W[4kk+2][n0+l-16], v1 = W[4kk+3][n0+l-16]
    #pragma unroll 4
    for (int i = 0; i < KDIM * NDIM; i += 256) {
        const int e  = i + t;          // coalesced global read
        const int k  = e >> 7;
        const int n  = e & 127;
        const int kk = k >> 2;
        const int kr = k & 3;
        const int lane = ((kr >> 1) << 4) | (n & 15);
        const int frag = (kk << 3) | (n >> 4);
        ldsW[(((frag << 6) + lane) << 1) + (kr & 1)] = w[e];
    }
    __syncthreads();

    const int wv   = t >> 5;           // wave id within block (0..7)
    const int lane = t & 31;
    const int row0 = blockIdx.x * 128 + wv * 16;

    // ---- A-fragment pointer (ISA 7.12.2, 32-bit A 16x4) ----
    //   lane l (<16):  {x[row0+l][4kk+0], x[row0+l][4kk+1]}
    //   lane l (>=16): {x[row0+l-16][4kk+2], x[row0+l-16][4kk+3]}
    const int rowA = row0 + (lane & 15);
    const float* aptr = x + (size_t)rowA * KDIM + ((lane >> 4) << 1);

    const v2f* __restrict__ bfrag = (const v2f*)ldsW;

    v8f acc[8];
    #pragma unroll
    for (int nt = 0; nt < 8; ++nt) acc[nt] = (v8f){0.f,0.f,0.f,0.f,0.f,0.f,0.f,0.f};

    // ---- K loop: 32 steps of K=4; 8 WMMAs per step (full N=128 reuse of A) ----
    #pragma unroll 2
    for (int kk = 0; kk < 32; ++kk) {
        const v2f a = *(const v2f*)(aptr + (kk << 2));   // global_load_b64
        #pragma unroll
        for (int nt = 0; nt < 8; ++nt) {
            const v2f b = bfrag[(((kk << 3) + nt) << 6) + lane];  // ds_load_b64, conflict-free
            acc[nt] = __builtin_amdgcn_wmma_f32_16x16x4_f32(
                false, a, false, b, (short)0, acc[nt], false, false);
        }
    }

    // ---- Bias + store (C/D layout: VGPR v -> M = v (+8 upper lanes), N = lane&15) ----
    const int ncol  = lane & 15;
    const int rbase = row0 + ((lane >> 4) << 3);
    #pragma unroll
    for (int nt = 0; nt < 8; ++nt) {
        const float bv = bias[nt * 16 + ncol];
        #pragma unroll
        for (int v = 0; v < 8; ++v) {
            y[(size_t)(rbase + v) * NDIM + nt * 16 + ncol] = acc[nt][v] + bv;
        }
    }
}

extern "C" void kernel_launch(void* const* d_in, const int* in_sizes, int n_in,
                              void* d_out, int out_size, void* d_ws, size_t ws_size,
                              hipStream_t stream) {
    const float* x    = (const float*)d_in[0];   // enc_x [262144,128]
    const float* w    = (const float*)d_in[1];   // weight [128,128]
    const float* bias = (const float*)d_in[2];   // bias [128]
    float* y          = (float*)d_out;           // [262144,128]

    dim3 grid(M_TOTAL / 128);
    dim3 block(256);
    hipLaunchKernelGGL(linear_wmma_f32_kernel, grid, block, 0, stream, x, w, bias, y);
}